// GIN_56745107915456
// MI455X (gfx1250) — compile-verified
//
#include <hip/hip_runtime.h>

typedef __attribute__((ext_vector_type(2))) float v2f;
typedef __attribute__((ext_vector_type(8))) float v8f;

// ---------------------------------------------------------------------------
// Fused GEMM: out[M,64] = relu( (in[M,K] @ W[K,64] + bias) * scale + shift )
// scale/shift implement BatchNorm1d (eval mode, running stats 0/1):
//   scale = gamma / sqrt(1 + eps), shift = beta   (or identity when null)
// WMMA f32 16x16x4, wave32. Block = 256 threads = 8 waves; each wave does a
// 16-row x 64-col strip (4 accumulators of v8f). W is staged transposed in
// LDS with stride K+4 so B-fragment v2f reads are LDS bank-conflict free.
// ---------------------------------------------------------------------------
template <int K>
__launch_bounds__(256)
__global__ void gemm64_bn_relu(const float* __restrict__ in,
                               const float* __restrict__ W,
                               const float* __restrict__ bias,
                               const float* __restrict__ gamma,
                               const float* __restrict__ beta,
                               float* __restrict__ out, int M) {
  constexpr int WS = K + 4;          // padded transpose stride
  __shared__ float WT[64 * WS];      // WT[n*WS + k] = W[k*64 + n]
  __shared__ float colS[64];
  __shared__ float colB[64];

  for (int i = threadIdx.x; i < K * 64; i += 256) {
    int k = i >> 6;
    int n = i & 63;
    WT[n * WS + k] = W[i];
  }
  if (threadIdx.x < 64) {
    int n = threadIdx.x;
    float s = 1.0f, b = 0.0f;
    if (gamma) s = gamma[n] * 0.9999950000374997f;  // 1/sqrt(1+1e-5)
    if (beta)  b = beta[n];
    colS[n] = s;
    colB[n] = bias[n] * s + b;
  }
  __syncthreads();

  const int wave = threadIdx.x >> 5;
  const int lane = threadIdx.x & 31;
  const int m16  = lane & 15;
  const int kh   = lane >> 4;                 // K-pair select (ISA A/B layout)
  const int row0 = blockIdx.x * 128 + wave * 16;

  int rowA = row0 + m16;
  if (rowA > M - 1) rowA = M - 1;             // clamp tail reads; stores guarded
  const float* __restrict__ arow = in + (size_t)rowA * K;

  v8f acc[4] = {v8f{}, v8f{}, v8f{}, v8f{}};

  for (int kk = 0; kk < K; kk += 4) {
    const int kbase = kk + 2 * kh;
    v2f a = *(const v2f*)(arow + kbase);      // A 16x4: lane half -> K pair
#pragma unroll
    for (int t = 0; t < 4; ++t) {
      // B 4x16: col = lane&15, lane half -> K pair; contiguous in WT
      v2f b = *(const v2f*)(&WT[(t * 16 + m16) * WS + kbase]);
      acc[t] = __builtin_amdgcn_wmma_f32_16x16x4_f32(
          false, a, false, b, (short)0, acc[t], false, false);
    }
  }

  // C/D layout: vgpr r holds row (lane<16 ? r : 8+r), col = lane&15
  const int rbase = kh * 8;
  const int ncol  = m16;
#pragma unroll
  for (int t = 0; t < 4; ++t) {
    int n   = t * 16 + ncol;
    float s = colS[n];
    float c = colB[n];
#pragma unroll
    for (int r = 0; r < 8; ++r) {
      int row = row0 + rbase + r;
      if (row < M) {
        float v = acc[t][r] * s + c;
        out[(size_t)row * 64 + n] = v > 0.0f ? v : 0.0f;
      }
    }
  }
}

// z = h  (float4 copy; the "(1+eps)*x" term of GIN with eps=0)
__global__ void copy_f4(const float4* __restrict__ in, float4* __restrict__ out,
                        int n) {
  int i = blockIdx.x * 256 + threadIdx.x;
  if (i < n) out[i] = in[i];
}

// z[dst[e], f] += h[src[e], f]   — 64 consecutive threads per edge: coalesced
// gather and coalesced atomic segment per edge; h/z are L2-resident (25.6 MB).
__global__ void scatter_add64(const float* __restrict__ h,
                              const int* __restrict__ src,
                              const int* __restrict__ dst,
                              float* __restrict__ z, int E) {
  int idx = blockIdx.x * 256 + threadIdx.x;
  int e = idx >> 6;
  if (e >= E) return;
  int f = idx & 63;
  int s = src[e];
  int d = dst[e];
  float v = h[(size_t)s * 64 + f];
  unsafeAtomicAdd(&z[(size_t)d * 64 + f], v);
}

__global__ void zero_f(float* __restrict__ p, int n) {
  int i = blockIdx.x * 256 + threadIdx.x;
  if (i < n) p[i] = 0.0f;
}

// per-graph sum pool (+ node counts via feat-0 lane)
__global__ void pool_add(const float* __restrict__ h,
                         const int* __restrict__ batch,
                         float* __restrict__ sums, float* __restrict__ cnt,
                         int N) {
  int idx = blockIdx.x * 256 + threadIdx.x;
  int n = idx >> 6;
  if (n >= N) return;
  int f = idx & 63;
  int g = batch[n];
  unsafeAtomicAdd(&sums[g * 64 + f], h[(size_t)n * 64 + f]);
  if (f == 0) unsafeAtomicAdd(&cnt[g], 1.0f);
}

// pooled = sums/max(cnt,1);  out = softmax(pooled @ Wl2 + bl2)   [128,10]
__global__ void head_kernel(const float* __restrict__ sums,
                            const float* __restrict__ cnt,
                            const float* __restrict__ Wl2,
                            const float* __restrict__ bl2,
                            float* __restrict__ out) {
  int g = threadIdx.x;  // 128 threads, 1 block
  float c = cnt[g];
  if (c < 1.0f) c = 1.0f;
  float inv = 1.0f / c;
  float p[10];
#pragma unroll
  for (int j = 0; j < 10; ++j) p[j] = bl2[j];
  for (int k = 0; k < 64; ++k) {
    float v = sums[g * 64 + k] * inv;
#pragma unroll
    for (int j = 0; j < 10; ++j) p[j] += v * Wl2[k * 10 + j];
  }
  float mx = p[0];
#pragma unroll
  for (int j = 1; j < 10; ++j) mx = fmaxf(mx, p[j]);
  float se = 0.0f;
#pragma unroll
  for (int j = 0; j < 10; ++j) {
    p[j] = __expf(p[j] - mx);
    se += p[j];
  }
  float r = 1.0f / se;
#pragma unroll
  for (int j = 0; j < 10; ++j) out[g * 10 + j] = p[j] * r;
}

extern "C" void kernel_launch(void* const* d_in, const int* in_sizes, int n_in,
                              void* d_out, int out_size, void* d_ws,
                              size_t ws_size, hipStream_t stream) {
  const float* x     = (const float*)d_in[0];
  const int*   ei    = (const int*)d_in[1];
  // d_in[2] = ew (unused by GIN forward)
  const int*   batch = (const int*)d_in[3];
  const float* W0    = (const float*)d_in[4];
  const float* b0    = (const float*)d_in[5];
  const float* cW1   = (const float*)d_in[6];
  const float* cb1   = (const float*)d_in[7];
  const float* cg1   = (const float*)d_in[8];
  const float* cbt1  = (const float*)d_in[9];
  const float* cW2   = (const float*)d_in[10];
  const float* cb2   = (const float*)d_in[11];
  const float* cg2   = (const float*)d_in[12];
  const float* cbt2  = (const float*)d_in[13];
  const float* Wl1   = (const float*)d_in[14];
  const float* bl1   = (const float*)d_in[15];
  const float* Wl2   = (const float*)d_in[16];
  const float* bl2   = (const float*)d_in[17];
  float* out = (float*)d_out;

  const int N = 100000;
  const int E = in_sizes[1] / 2;  // ei is [2,E]
  const int G = 128;
  const int* src = ei;
  const int* dst = ei + E;

  char* ws = (char*)d_ws;
  const size_t HBYTES = (size_t)N * 64 * sizeof(float);  // 25.6 MB
  float* hA   = (float*)ws;
  float* hB   = (float*)(ws + HBYTES);
  float* sums = (float*)(ws + 2 * HBYTES);  // [G,64]
  float* cnt  = sums + G * 64;              // [G]

  const int gGemm = (N + 127) / 128;

  // h = relu(x @ W0 + b0)
  gemm64_bn_relu<128><<<gGemm, 256, 0, stream>>>(x, W0, b0, nullptr, nullptr,
                                                 hA, N);

  for (int i = 0; i < 4; ++i) {
    // z = h + sum_{j in N(i)} h_j
    copy_f4<<<(N * 64 / 4 + 255) / 256, 256, 0, stream>>>((const float4*)hA,
                                                          (float4*)hB,
                                                          N * 64 / 4);
    scatter_add64<<<(E * 64 + 255) / 256, 256, 0, stream>>>(hA, src, dst, hB, E);
    // z2 = relu(bn(z @ W1 + b1));  h = relu(bn(z2 @ W2 + b2))
    gemm64_bn_relu<64><<<gGemm, 256, 0, stream>>>(
        hB, cW1 + i * 64 * 64, cb1 + i * 64, cg1 + i * 64, cbt1 + i * 64, hA, N);
    gemm64_bn_relu<64><<<gGemm, 256, 0, stream>>>(
        hA, cW2 + i * 64 * 64, cb2 + i * 64, cg2 + i * 64, cbt2 + i * 64, hB, N);
    float* t = hA; hA = hB; hB = t;  // hA = new h
  }

  // h = relu(h @ Wl1 + bl1)
  gemm64_bn_relu<64><<<gGemm, 256, 0, stream>>>(hA, Wl1, bl1, nullptr, nullptr,
                                                hB, N);

  zero_f<<<(G * 64 + G + 255) / 256, 256, 0, stream>>>(sums, G * 64 + G);
  pool_add<<<(N * 64 + 255) / 256, 256, 0, stream>>>(hB, batch, sums, cnt, N);
  head_kernel<<<1, 128, 0, stream>>>(sums, cnt, Wl2, bl2, out);
}